// GCN_CUSTOM_7722351198605
// MI455X (gfx1250) — compile-verified
//
#include <hip/hip_runtime.h>
#include <math.h>

#define N_NODES 50000
#define DIM     128
#define NCLS    40

typedef float v2f __attribute__((ext_vector_type(2)));
typedef float v8f __attribute__((ext_vector_type(8)));

// ---------------- degree / normalization ----------------
__global__ void k_fill_deg(float* deg, int n) {
    int i = blockIdx.x * blockDim.x + threadIdx.x;
    if (i < n) deg[i] = 1.0f;                       // self-loop contributes 1
}

__global__ void k_count(const int* __restrict__ dst, float* __restrict__ deg, int e) {
    int i = blockIdx.x * blockDim.x + threadIdx.x;
    if (i < e) atomicAdd(&deg[dst[i]], 1.0f);
}

__global__ void k_dinv(float* deg, int n) {
    int i = blockIdx.x * blockDim.x + threadIdx.x;
    if (i < n) {
        float d = deg[i];
        deg[i] = (d > 0.0f) ? rsqrtf(d) : 0.0f;     // dinv stored in place
    }
}

// ---------------- fp32 WMMA GEMM: H = X @ W^T  (N x 128 @ 128 x 128) ----------------
// One wave per 16x16 output tile. A(16x4 f32): lanes0-15 K={0,1}, lanes16-31 K={2,3}.
// B(4x16): v0 = Wt[k+2*kh, col], v1 = Wt[k+2*kh+1, col] with Wt[k,c] = W[c,k].
__global__ void k_gemm128(const float* __restrict__ X, const float* __restrict__ W,
                          float* __restrict__ H) {
    const int lane = threadIdx.x & 31;
    const int lm   = lane & 15;
    const int kh   = lane >> 4;                     // 0 or 1
    const int row0 = blockIdx.x * 16;
    const int col0 = blockIdx.y * 16;

    const float* arow = X + (size_t)(row0 + lm) * DIM + 2 * kh;
    const float* brow = W + (size_t)(col0 + lm) * DIM + 2 * kh;  // W row = output col

    v8f acc = {0.f, 0.f, 0.f, 0.f, 0.f, 0.f, 0.f, 0.f};
#pragma unroll 4
    for (int k = 0; k < DIM; k += 4) {
        v2f a = *(const v2f*)(arow + k);
        v2f b = *(const v2f*)(brow + k);
        acc = __builtin_amdgcn_wmma_f32_16x16x4_f32(false, a, false, b,
                                                    (short)0, acc, false, false);
    }

    float* out = H + (size_t)(row0 + 8 * kh) * DIM + col0 + lm;
#pragma unroll
    for (int r = 0; r < 8; ++r)
        out[(size_t)r * DIM] = acc[r];
}

// ---------------- aggregation ----------------
// agg[i,:] = h[i,:] * dinv[i]^2   (self-loop term; also initializes the buffer)
__global__ void k_selfinit(const float* __restrict__ H, const float* __restrict__ dinv,
                           float* __restrict__ A, size_t total) {
    size_t idx = (size_t)blockIdx.x * blockDim.x + threadIdx.x;
    if (idx < total) {
        float di = dinv[idx >> 7];
        A[idx] = H[idx] * di * di;
    }
}

// one wave per edge: lanes cover the 128-float row as float4 each
__global__ void k_scatter(const int* __restrict__ src, const int* __restrict__ dst,
                          const float* __restrict__ dinv, const float* __restrict__ H,
                          float* __restrict__ A, int e) {
    int w    = (int)(((size_t)blockIdx.x * blockDim.x + threadIdx.x) >> 5);
    int lane = threadIdx.x & 31;
    if (w >= e) return;
    int s = src[w], d = dst[w];
    float nrm = dinv[s] * dinv[d];
    const float4 v = *(const float4*)(H + (size_t)s * DIM + lane * 4);
    float* o = A + (size_t)d * DIM + lane * 4;
    atomicAdd(o + 0, v.x * nrm);
    atomicAdd(o + 1, v.y * nrm);
    atomicAdd(o + 2, v.z * nrm);
    atomicAdd(o + 3, v.w * nrm);
}

__global__ void k_bias_relu(float* __restrict__ A, const float* __restrict__ b,
                            size_t total) {
    size_t idx = (size_t)blockIdx.x * blockDim.x + threadIdx.x;
    if (idx < total) {
        float v = A[idx] + b[idx & (DIM - 1)];
        A[idx] = v > 0.0f ? v : 0.0f;
    }
}

// ---------------- head: logits = X @ Wl^T + bl  (cols padded 40 -> 48) ----------------
__global__ void k_head(const float* __restrict__ X, const float* __restrict__ Wl,
                       const float* __restrict__ bl, float* __restrict__ out) {
    const int lane = threadIdx.x & 31;
    const int lm   = lane & 15;
    const int kh   = lane >> 4;
    const int row0 = blockIdx.x * 16;
    const int col0 = blockIdx.y * 16;
    const int col  = col0 + lm;
    const bool colok = col < NCLS;

    const float* arow = X + (size_t)(row0 + lm) * DIM + 2 * kh;
    const float* brow = Wl + (size_t)(colok ? col : 0) * DIM + 2 * kh;

    v8f acc = {0.f, 0.f, 0.f, 0.f, 0.f, 0.f, 0.f, 0.f};
#pragma unroll 4
    for (int k = 0; k < DIM; k += 4) {
        v2f a = *(const v2f*)(arow + k);
        v2f b = *(const v2f*)(brow + k);
        if (!colok) { b.x = 0.f; b.y = 0.f; }       // lanes beyond 40 contribute zeros
        acc = __builtin_amdgcn_wmma_f32_16x16x4_f32(false, a, false, b,
                                                    (short)0, acc, false, false);
    }

    if (colok) {
        float bb = bl[col];
        float* o = out + (size_t)(row0 + 8 * kh) * NCLS + col;
#pragma unroll
        for (int r = 0; r < 8; ++r)
            o[(size_t)r * NCLS] = acc[r] + bb;
    }
}

// ---------------- log-softmax, one wave32 per row of 40 ----------------
__global__ void k_logsoftmax(float* __restrict__ out, int n) {
    int row  = blockIdx.x * (blockDim.x >> 5) + (threadIdx.x >> 5);
    int lane = threadIdx.x & 31;
    if (row >= n) return;
    float* p = out + (size_t)row * NCLS;

    float x0 = p[lane];                                   // lanes 0..31 all < 40
    float x1 = (lane < NCLS - 32) ? p[lane + 32] : -INFINITY;

    float m = fmaxf(x0, x1);
    for (int o = 16; o > 0; o >>= 1) m = fmaxf(m, __shfl_xor(m, o, 32));

    float s = expf(x0 - m) + ((lane < NCLS - 32) ? expf(x1 - m) : 0.0f);
    for (int o = 16; o > 0; o >>= 1) s += __shfl_xor(s, o, 32);

    float L = logf(s);
    p[lane] = x0 - m - L;
    if (lane < NCLS - 32) p[lane + 32] = x1 - m - L;
}

// ---------------- driver ----------------
extern "C" void kernel_launch(void* const* d_in, const int* in_sizes, int n_in,
                              void* d_out, int out_size, void* d_ws, size_t ws_size,
                              hipStream_t stream) {
    const float* x  = (const float*)d_in[0];
    const int*   ei = (const int*)d_in[1];           // (2, E) int32
    const float* W1 = (const float*)d_in[2];
    const float* b1 = (const float*)d_in[3];
    const float* W2 = (const float*)d_in[4];
    const float* b2 = (const float*)d_in[5];
    const float* Wl = (const float*)d_in[6];
    const float* bl = (const float*)d_in[7];

    const int n = in_sizes[0] / DIM;                 // 50000
    const int e = in_sizes[1] / 2;                   // 600000
    const int* src = ei;
    const int* dst = ei + e;
    float* outp = (float*)d_out;

    // workspace carve-up: dinv[n] | bufA[n*128] | bufB[n*128]  (~51.5 MB)
    float* dinv = (float*)d_ws;
    float* bufA = dinv + n;
    float* bufB = bufA + (size_t)n * DIM;

    const size_t total = (size_t)n * DIM;
    const int ew_grid   = (int)((total + 255) / 256);
    const int n_grid    = (n + 255) / 256;
    const int e_grid    = (e + 255) / 256;
    const int scat_grid = (int)(((size_t)e * 32 + 255) / 256);
    const dim3 gemm_grid(n / 16, DIM / 16);
    const dim3 head_grid(n / 16, (NCLS + 15) / 16);

    // degrees -> dinv
    k_fill_deg<<<n_grid, 256, 0, stream>>>(dinv, n);
    k_count   <<<e_grid, 256, 0, stream>>>(dst, dinv, e);
    k_dinv    <<<n_grid, 256, 0, stream>>>(dinv, n);

    // layer 1: h = x @ W1^T ; agg ; +b1 ; relu
    k_gemm128  <<<gemm_grid, 32, 0, stream>>>(x, W1, bufA);
    k_selfinit <<<ew_grid, 256, 0, stream>>>(bufA, dinv, bufB, total);
    k_scatter  <<<scat_grid, 256, 0, stream>>>(src, dst, dinv, bufA, bufB, e);
    k_bias_relu<<<ew_grid, 256, 0, stream>>>(bufB, b1, total);

    // layer 2
    k_gemm128  <<<gemm_grid, 32, 0, stream>>>(bufB, W2, bufA);
    k_selfinit <<<ew_grid, 256, 0, stream>>>(bufA, dinv, bufB, total);
    k_scatter  <<<scat_grid, 256, 0, stream>>>(src, dst, dinv, bufA, bufB, e);
    k_bias_relu<<<ew_grid, 256, 0, stream>>>(bufB, b2, total);

    // head + log-softmax
    k_head      <<<head_grid, 32, 0, stream>>>(bufB, Wl, bl, outp);
    k_logsoftmax<<<(n + 7) / 8, 256, 0, stream>>>(outp, n);
}